// WhdrHingeLoss_15994458211239
// MI455X (gfx1250) — compile-verified
//
#include <hip/hip_runtime.h>
#include <hip/hip_bf16.h>

// ---------------- vector types ----------------
typedef __attribute__((ext_vector_type(2))) float v2f;
typedef __attribute__((ext_vector_type(8))) float v8f;
typedef __attribute__((ext_vector_type(4))) int   v4i;
typedef __attribute__((ext_vector_type(4))) float v4f;

// ---------------- CDNA5 wave32 reduction via V_WMMA_F32_16X16X4_F32 ----------
// Sum the per-lane float across the 32-lane wave. Returns total in all lanes.
// A-matrix 16x4 f32 layout: lanes 0-15 VGPR0 = A[m,0]; lanes 16-31 VGPR0 = A[m,2].
// With A VGPR1 = 0 and B = ones(4x16): D[m,n] = p_m + p_{m+16}.
// D layout: VGPR j, lanes 0-15 -> D[j, lane]; lanes 16-31 -> D[8+j, lane-16].
// So (sum of my 8 D regs) + (same from lane^16) = full 32-lane total.
__device__ __forceinline__ float wave_sum_wmma(float x) {
    v2f a; a.x = x; a.y = 0.0f;
    v2f b; b.x = 1.0f; b.y = 1.0f;
    v8f c = {};
    v8f d = __builtin_amdgcn_wmma_f32_16x16x4_f32(
        /*neg_a=*/false, a, /*neg_b=*/false, b,
        /*c_mod=*/(short)0, c, /*reuse_a=*/false, /*reuse_b=*/false);
    float s = (d[0] + d[1]) + (d[2] + d[3]) + ((d[4] + d[5]) + (d[6] + d[7]));
    s += __shfl_xor(s, 16, 32);
    return s;
}

// ---------------- constants ----------------
#define WHDR_EPS   1e-10f
#define WHDR_B     1.15f            // 1 + DELTA + MARGIN
#define WHDR_INVB  (1.0f / 1.15f)
#define WHDR_BR    1.05f            // 1 + DELTA - MARGIN
#define WHDR_BL    (1.0f / 1.05f)
#define LUM_K0     (255.0f / 768.0f)
#define LUM_K1     (1.0f / 256.0f)

// ---------------- stage 1: build luminance table ----------------
__global__ void whdr_build_v(const float* __restrict__ vin,
                             float* __restrict__ vtab, int hw) {
    int i = blockIdx.x * blockDim.x + threadIdx.x;
    if (i < hw) {
        float s = vin[i] + vin[i + hw] + vin[i + 2 * hw];
        vtab[i] = s * LUM_K0 + LUM_K1;
    }
}

// ---------------- stage 2: main streaming + gather + per-block reduce -------
__device__ __forceinline__ float whdr_term(float r1, float r2, int dk, float wt) {
    float ratio = r1 / (r2 + WHDR_EPS);
    float rinv  = r2 / (r1 + WHDR_EPS);
    float l1 = (ratio > WHDR_INVB) ? (ratio - WHDR_INVB) + (WHDR_B - rinv) : 0.0f;
    float l2 = (ratio < WHDR_B)    ? (WHDR_B - ratio) + (rinv - WHDR_INVB) : 0.0f;
    float l0 = (ratio > WHDR_BR)   ? (ratio - WHDR_BR) + (WHDR_BL - rinv)
             : ((ratio < WHDR_BL)  ? (WHDR_BL - ratio) + (rinv - WHDR_BR) : 0.0f);
    float per = (dk == 1) ? l1 : ((dk == 2) ? l2 : l0);
    return wt * per;
}

template <bool USE_TABLE>
__global__ void whdr_main(const float* __restrict__ vtab,
                          const float* __restrict__ ch0,
                          const float* __restrict__ ch1,
                          const float* __restrict__ ch2,
                          const v4i*  __restrict__ coords4,  // one v4i per element
                          const v4i*  __restrict__ darker4,  // 4 elements per v4i
                          const v4f*  __restrict__ wts4,     // 4 elements per v4f
                          float* __restrict__ partials,
                          int numGroups) {
    __shared__ float lds[8];
    const int stride = gridDim.x * blockDim.x;
    float acc = 0.0f;

    for (int g = blockIdx.x * blockDim.x + threadIdx.x; g < numGroups; g += stride) {
        // prefetch next chunk of coords (speculative; dropped if OOB)
        __builtin_prefetch((const void*)(coords4 + 4 * (long)(g + stride)), 0, 1);

        // Streamed-once data: non-temporal loads so the gather table stays
        // resident in WGP$/L2 while ~200 MB flows past at HBM rate.
        v4i dd = __builtin_nontemporal_load(&darker4[g]);
        v4f ww = __builtin_nontemporal_load(&wts4[g]);
        const v4i* cg = coords4 + 4 * (long)g;

#pragma unroll
        for (int j = 0; j < 4; ++j) {
            v4i cc = __builtin_nontemporal_load(&cg[j]);
            int idx1 = (cc[1] << 9) | cc[0];   // y1*512 + x1, both in [0,512)
            int idx2 = (cc[3] << 9) | cc[2];   // y2*512 + x2
            float r1, r2;
            if (USE_TABLE) {
                r1 = vtab[idx1];               // temporal: table is hot
                r2 = vtab[idx2];
            } else {
                r1 = (ch0[idx1] + ch1[idx1] + ch2[idx1]) * LUM_K0 + LUM_K1;
                r2 = (ch0[idx2] + ch1[idx2] + ch2[idx2]) * LUM_K0 + LUM_K1;
            }
            acc += whdr_term(r1, r2, dd[j], ww[j]);
        }
    }

    // wave32 reduction through the matrix pipe (EXEC is all-ones here)
    float wsum = wave_sum_wmma(acc);
    int wave = threadIdx.x >> 5;
    int lane = threadIdx.x & 31;
    if (lane == 0) lds[wave] = wsum;
    __syncthreads();
    if (threadIdx.x == 0) {
        float bsum = 0.0f;
#pragma unroll
        for (int i = 0; i < 8; ++i) bsum += lds[i];
        partials[blockIdx.x] = bsum;
    }
}

// ---------------- stage 3: deterministic final reduction (one wave) ---------
__global__ void whdr_final(const float* __restrict__ partials, int n,
                           float* __restrict__ out, float invN) {
    int lane = threadIdx.x; // 32 threads, one full wave
    float p = 0.0f;
    for (int i = lane; i < n; i += 32) p += partials[i];
    float tot = wave_sum_wmma(p);
    if (lane == 0) out[0] = tot * invN;
}

// ---------------- host launcher ----------------
extern "C" void kernel_launch(void* const* d_in, const int* in_sizes, int n_in,
                              void* d_out, int out_size, void* d_ws, size_t ws_size,
                              hipStream_t stream) {
    const float* v_input = (const float*)d_in[0]; // (3, 512, 512)
    const int*   coords  = (const int*)d_in[1];   // (N, 4)
    const int*   darker  = (const int*)d_in[2];   // (N,)
    const float* weights = (const float*)d_in[3]; // (N,)
    float*       out     = (float*)d_out;

    const int N  = in_sizes[2];
    const int hw = in_sizes[0] / 3;               // 512*512

    const int BLOCK = 256;
    const int numGroups = N / 4;                  // N is a multiple of 4 here
    int blocksB = (numGroups + BLOCK - 1) / BLOCK;
    if (blocksB > 2048) blocksB = 2048;           // ~16 elements per thread
    if (blocksB < 1) blocksB = 1;

    const size_t tabBytes  = (size_t)hw * sizeof(float);
    const size_t needTable = tabBytes + (size_t)blocksB * sizeof(float);
    const bool useTable = (ws_size >= needTable);

    const v4i* coords4 = (const v4i*)coords;
    const v4i* darker4 = (const v4i*)darker;
    const v4f* wts4    = (const v4f*)weights;
    const float invN   = 1.0f / (float)N;

    if (useTable) {
        float* vtab     = (float*)d_ws;
        float* partials = (float*)((char*)d_ws + tabBytes);
        whdr_build_v<<<(hw + BLOCK - 1) / BLOCK, BLOCK, 0, stream>>>(v_input, vtab, hw);
        whdr_main<true><<<blocksB, BLOCK, 0, stream>>>(
            vtab, nullptr, nullptr, nullptr,
            coords4, darker4, wts4, partials, numGroups);
        whdr_final<<<1, 32, 0, stream>>>(partials, blocksB, out, invN);
    } else {
        // fallback: gather the 3 channels directly (3 MB working set, still L2-resident)
        if ((size_t)blocksB * sizeof(float) > ws_size)
            blocksB = (int)(ws_size / sizeof(float)) > 0 ? (int)(ws_size / sizeof(float)) : 1;
        float* partials = (float*)d_ws;
        whdr_main<false><<<blocksB, BLOCK, 0, stream>>>(
            nullptr, v_input, v_input + hw, v_input + 2 * hw,
            coords4, darker4, wts4, partials, numGroups);
        whdr_final<<<1, 32, 0, stream>>>(partials, blocksB, out, invN);
    }
}